// PromptClip_42984032698800
// MI455X (gfx1250) — compile-verified
//
#include <hip/hip_runtime.h>
#include <math.h>

typedef __attribute__((ext_vector_type(2))) float v2f;
typedef __attribute__((ext_vector_type(4))) float v4f;
typedef __attribute__((ext_vector_type(8))) float v8f;

#define COS_EPS 1e-8f
#define KDIM    512
#define PDIM    768
#define NSEL    32
#define NEG_BIG -1e30f

// ---------------------------------------------------------------------------
// Kernel 1: per-row dot(prompt_k[r], x) via V_WMMA_F32_16X16X4_F32 plus row
// sum-of-squares, writing dist[r] = dot / max(||k_r||, eps).
// One wave handles a 16-row tile; block = 64 threads = 2 waves = 32 rows.
//
// K-chunking: per chunk of 8 K-values, each lane does ONE b128 load:
//   low lanes (0-15)  hold K = {k0..k0+3}   of their row,
//   high lanes (16-31) hold K = {k0+4..k0+7}.
// WMMA #1 consumes slots (x,y) of both halves, WMMA #2 consumes (z,w).
// Since the dot is a plain sum over K and A/B use the identical K mapping,
// the accumulated result equals the true dot product.
// ---------------------------------------------------------------------------
__global__ __launch_bounds__(64) void cosine_dot_kernel(
    const float* __restrict__ x, const float* __restrict__ K,
    float* __restrict__ dist, int N)
{
    const int lane    = threadIdx.x & 31;
    const int wave    = threadIdx.x >> 5;
    const int rowBase = blockIdx.x * 32 + wave * 16;

    const int m      = lane & 15;   // A-matrix row owned by this lane
    const int halfId = lane >> 4;   // 0 -> K quad {0..3}, 1 -> K quad {4..7}

    // Clamp load row for safety on ragged N (store is guarded below).
    int loadRow = rowBase + m;
    if (loadRow > N - 1) loadRow = N - 1;

    const float* kA = K + (size_t)loadRow * KDIM + 4 * halfId;  // 16B aligned
    const float* xB = x + 4 * halfId;                           // 16B aligned

    // Prefetch the back half of this lane's row stream (global_prefetch_b8).
    __builtin_prefetch(kA + 256, 0, 1);

    v8f   acc   = {};
    float sumsq = 0.0f;

    #pragma unroll 2
    for (int k0 = 0; k0 < KDIM; k0 += 8) {
        v4f a = *(const v4f*)(kA + k0);   // global_load_b128
        v4f b = *(const v4f*)(xB + k0);   // global_load_b128

        sumsq = fmaf(a.x, a.x, sumsq);
        sumsq = fmaf(a.y, a.y, sumsq);
        sumsq = fmaf(a.z, a.z, sumsq);
        sumsq = fmaf(a.w, a.w, sumsq);

        v2f a0; a0.x = a.x; a0.y = a.y;
        v2f a1; a1.x = a.z; a1.y = a.w;
        v2f b0; b0.x = b.x; b0.y = b.y;
        v2f b1; b1.x = b.z; b1.y = b.w;

        // D = A*B + C   (f32 in / f32 accumulate)
        acc = __builtin_amdgcn_wmma_f32_16x16x4_f32(
            false, a0, false, b0, (short)0, acc, false, false);
        acc = __builtin_amdgcn_wmma_f32_16x16x4_f32(
            false, a1, false, b1, (short)0, acc, false, false);
    }

    // Full ||row||^2: lane l holds the {0..3 mod 8} partial, lane l^16 the
    // {4..7 mod 8} partial of the same row.
    sumsq += __shfl_xor(sumsq, 16, 32);

    // Extract dots from the C/D layout: VGPR r holds row r (lanes 0-15) and
    // row r+8 (lanes 16-31); every column is identical (B replicated).
    const int r = lane & 7;
    float sel = acc[0];
    if (r == 1) sel = acc[1];
    if (r == 2) sel = acc[2];
    if (r == 3) sel = acc[3];
    if (r == 4) sel = acc[4];
    if (r == 5) sel = acc[5];
    if (r == 6) sel = acc[6];
    if (r == 7) sel = acc[7];
    const float selX = __shfl_xor(sel, 16, 32);   // pull rows 8..15 into lanes 8..15

    const int outRow = rowBase + lane;            // lanes 0..15 own rows 0..15
    if (lane < 16 && outRow < N) {
        const float dot = (lane < 8) ? sel : selX;
        const float nk  = sqrtf(sumsq);
        dist[outRow] = dot / fmaxf(nk, COS_EPS);
    }
}

// ---------------------------------------------------------------------------
// Kernel 2: single-block iterative top-32 argmax over dist[N].
// Records topVal (scaled by 1/max(||x||,eps) to match the reference cosine)
// and topIdx, kills selected entries, and writes score = mean(topVal).
// dist[] is fully rewritten by kernel 1 every call -> graph-replay safe.
// ---------------------------------------------------------------------------
__global__ __launch_bounds__(1024) void topk_kernel(
    const float* __restrict__ x, float* __restrict__ dist,
    float* __restrict__ topVal, int* __restrict__ topIdx,
    float* __restrict__ scoreOut, int N)
{
    __shared__ float sVal[1024];
    __shared__ int   sIdx[1024];
    __shared__ float sNxInv;

    const int tid = threadIdx.x;

    // ||x|| reduction
    float xv = (tid < KDIM) ? x[tid] : 0.0f;
    sVal[tid] = xv * xv;
    __syncthreads();
    for (int s = 512; s > 0; s >>= 1) {
        if (tid < s) sVal[tid] += sVal[tid + s];
        __syncthreads();
    }
    if (tid == 0) sNxInv = 1.0f / fmaxf(sqrtf(sVal[0]), COS_EPS);
    __syncthreads();
    const float nxInv = sNxInv;

    float scoreAcc = 0.0f;

    for (int i = 0; i < NSEL; ++i) {
        // per-thread scan (ascending j + strict '>' keeps lowest index on ties)
        float best = NEG_BIG;
        int   bi   = 0x7fffffff;
        for (int j = tid; j < N; j += 1024) {
            float d = dist[j];
            if (d > best) { best = d; bi = j; }
        }
        sVal[tid] = best; sIdx[tid] = bi;
        __syncthreads();
        for (int s = 512; s > 0; s >>= 1) {
            if (tid < s) {
                float ov = sVal[tid + s]; int oi = sIdx[tid + s];
                if (ov > sVal[tid] || (ov == sVal[tid] && oi < sIdx[tid])) {
                    sVal[tid] = ov; sIdx[tid] = oi;
                }
            }
            __syncthreads();
        }
        if (tid == 0) {
            const int   win = sIdx[0];
            const float tv  = sVal[0] * nxInv;
            topIdx[i] = win;
            topVal[i] = tv;
            scoreAcc += tv;
            dist[win] = NEG_BIG;     // exclude from next round
        }
        __syncthreads();             // make dist[win] update visible
    }

    if (tid == 0) scoreOut[0] = scoreAcc * (1.0f / NSEL);
}

// ---------------------------------------------------------------------------
// Kernel 3: gather the 32 selected prompt_v rows into d_out.
// ---------------------------------------------------------------------------
__global__ __launch_bounds__(256) void gather_kernel(
    const float* __restrict__ V, const int* __restrict__ topIdx,
    float* __restrict__ out)
{
    const int b   = blockIdx.x;
    const int row = topIdx[b];
    const float* __restrict__ src = V + (size_t)row * PDIM;
    float* __restrict__ dst = out + (size_t)b * PDIM;
    for (int c = threadIdx.x; c < PDIM; c += 256) dst[c] = src[c];
}

// ---------------------------------------------------------------------------
extern "C" void kernel_launch(void* const* d_in, const int* in_sizes, int n_in,
                              void* d_out, int out_size, void* d_ws, size_t ws_size,
                              hipStream_t stream)
{
    const float* x  = (const float*)d_in[0];   // [1,512]
    const float* pk = (const float*)d_in[1];   // [N,512]
    const float* pv = (const float*)d_in[2];   // [N,768]
    (void)n_in; (void)out_size; (void)ws_size;

    const int N = in_sizes[1] / KDIM;          // 100000

    // Workspace layout
    float* dist   = (float*)d_ws;              // N floats
    float* topVal = dist + N;                  // 32 floats
    int*   topIdx = (int*)(topVal + NSEL);     // 32 ints

    float* out      = (float*)d_out;           // [32,768] then score
    float* scoreOut = out + NSEL * PDIM;

    const int tiles = (N + 31) / 32;           // 32 rows per 64-thread block
    cosine_dot_kernel<<<tiles, 64, 0, stream>>>(x, pk, dist, N);
    topk_kernel<<<1, 1024, 0, stream>>>(x, dist, topVal, topIdx, scoreOut, N);
    gather_kernel<<<NSEL, 256, 0, stream>>>(pv, topIdx, out);
}